// BeliefPlausibility_35656818492190
// MI455X (gfx1250) — compile-verified
//
#include <hip/hip_runtime.h>

typedef float v2f __attribute__((ext_vector_type(2)));
typedef float v8f __attribute__((ext_vector_type(8)));

// D = A(16x4 f32) * B(4x16 f32) + C(16x16 f32), exact f32 WMMA on gfx1250.
__device__ __forceinline__ v8f wmma_f32_4(v2f a, v2f b, v8f c) {
    return __builtin_amdgcn_wmma_f32_16x16x4_f32(
        /*neg_a=*/false, a, /*neg_b=*/false, b,
        /*c_mod=*/(short)0, c, /*reuse_a=*/false, /*reuse_b=*/false);
}

// One wave handles 16 pixels: bel/pl over all 64 subsets via 12 WMMAs.
__global__ __launch_bounds__(256) void belpl_wmma_kernel(
        const float* __restrict__ in, float* __restrict__ out_bel,
        float* __restrict__ out_pl, int ntiles) {
    const int lane  = threadIdx.x & 31;
    const int wave  = blockIdx.x * (blockDim.x >> 5) + (threadIdx.x >> 5);
    if (wave >= ntiles) return;                 // wave-uniform exit (EXEC stays all-1s)

    const long long base = (long long)wave * 16;   // first pixel of this tile
    const int  ncol  = lane & 15;                  // N column / pixel row id
    const int  khalf = lane >> 4;                  // 0: K={0,1}, 1: K={2,3}

    // ---- A fragments: ISA 32-bit A 16x4 layout ----
    // lanes 0-15: M=lane, V0=K0, V1=K1 ; lanes 16-31: M=lane-16, V0=K2, V1=K3
    const float* row = in + (base + ncol) * 7;
    v2f a1, a2;
    a1.x = khalf ? row[2] : row[0];
    a1.y = khalf ? row[3] : row[1];
    a2.x = khalf ? row[6] : row[4];   // K=6 slot carries m(Omega)
    a2.y = khalf ? 0.0f   : row[5];   // K=7 slot = pad

    // ---- B fragments from subset bit masks (mirrored K split) ----
    // VGPR v, lane l: K = v + 2*khalf, N = l&15 ; subset s = 16*t + N
    v2f b1[4], b2b[4], b2p[4];
#pragma unroll
    for (int t = 0; t < 4; ++t) {
        const int s  = 16 * t + ncol;
        const int K0 = 2 * khalf;          // 0 or 2
        const int K1 = K0 + 1;             // 1 or 3
        b1[t].x = (float)((s >> K0) & 1);  // channels 0..3
        b1[t].y = (float)((s >> K1) & 1);
        const int e0 = 4 + K0;             // 4 or 6
        const int e1 = 4 + K1;             // 5 or 7
        const float bit_e0 = (e0 < 6) ? (float)((s >> e0) & 1) : 0.0f;
        const float bit_e1 = (e1 < 6) ? (float)((s >> e1) & 1) : 0.0f;
        b2b[t].x = bit_e0;                 // bel: omega row = 0
        b2b[t].y = bit_e1;
        b2p[t].x = (e0 == 6) ? ((s >= 1 && s <= 62) ? 1.0f : 0.0f) : bit_e0;
        b2p[t].y = bit_e1;                 // pl: omega row = 1 on proper subsets
    }

    const v8f zero = {};
#pragma unroll
    for (int t = 0; t < 4; ++t) {
        v8f c0 = wmma_f32_4(a1, b1[t],  zero);  // channels 0..3
        v8f cb = wmma_f32_4(a2, b2b[t], c0);    // + channels 4,5        -> bel
        v8f cp = wmma_f32_4(a2, b2p[t], c0);    // + channels 4,5,omega  -> pl

        if (t == 3 && ncol == 15) {             // subset 63 := 1 exactly
#pragma unroll
            for (int r = 0; r < 8; ++r) { cb[r] = 1.0f; cp[r] = 1.0f; }
        }

        // D layout: VGPR r, this lane -> pixel base+r+8*khalf, subset 16t+ncol.
        // Half-wave writes a contiguous 64B run; NT hint (write-once stream).
#pragma unroll
        for (int r = 0; r < 8; ++r) {
            const long long p = base + r + 8 * khalf;
            const long long o = p * 64 + 16 * t + ncol;
            __builtin_nontemporal_store(cb[r], out_bel + o);
            __builtin_nontemporal_store(cp[r], out_pl  + o);
        }
    }
}

// Scalar tail for pixel counts not divisible by 16 (unused for the given shape).
__global__ void belpl_tail_kernel(const float* __restrict__ in,
                                  float* __restrict__ out_bel,
                                  float* __restrict__ out_pl,
                                  long long start_pix, long long npix) {
    long long idx = start_pix * 64 + (long long)blockIdx.x * blockDim.x + threadIdx.x;
    if (idx >= npix * 64) return;
    const long long p = idx >> 6;
    const int s = (int)(idx & 63);
    const float* r = in + p * 7;
    float bel, pl;
    if (s == 0)       { bel = 0.0f; pl = 0.0f; }
    else if (s == 63) { bel = 1.0f; pl = 1.0f; }
    else {
        bel = 0.0f;
#pragma unroll
        for (int c = 0; c < 6; ++c)
            if ((s >> c) & 1) bel += r[c];
        pl = bel + r[6];
    }
    out_bel[idx] = bel;
    out_pl[idx]  = pl;
}

extern "C" void kernel_launch(void* const* d_in, const int* in_sizes, int n_in,
                              void* d_out, int out_size, void* d_ws, size_t ws_size,
                              hipStream_t stream) {
    const float* in = (const float*)d_in[0];
    float* out = (float*)d_out;

    const long long P = (long long)in_sizes[0] / 7;   // 1,916,928 pixels
    float* out_bel = out;
    float* out_pl  = out + P * 64;                    // outputs concatenated flat

    const int ntiles = (int)(P / 16);
    if (ntiles > 0) {
        const int wavesPerBlock = 8;                  // 256 threads, wave32
        const int blocks = (ntiles + wavesPerBlock - 1) / wavesPerBlock;
        belpl_wmma_kernel<<<blocks, 256, 0, stream>>>(in, out_bel, out_pl, ntiles);
    }
    const long long done = (long long)ntiles * 16;
    const long long rem  = P - done;
    if (rem > 0) {
        const long long work = rem * 64;
        const int tb = 256;
        const int gb = (int)((work + tb - 1) / tb);
        belpl_tail_kernel<<<gb, tb, 0, stream>>>(in, out_bel, out_pl, done, P);
    }
}